// GainSuppressionMethod_24120536334968
// MI455X (gfx1250) — compile-verified
//
#include <hip/hip_runtime.h>
#include <cstdint>
#include <cstddef>

// ---------------- problem constants (from reference) ----------------
#define NF      257
#define NT_T    4000
#define NBLK_F  11          // frequency chunks per batch (24 output bins each)
#define OUTB    24          // 12 output bins per 16-lane half, 2-bin halo each side
#define TT      128         // time tile (steps per TDM tile)
#define NTILE   32          // ceil(4000/128): last tile has 32 valid steps
#define ROWF    132         // floats per LDS row: 128 + 4 pad (TDM pad feature)
#define ROWS_IN 28          // contiguous bin rows loaded per tile (24 + 2x2 halo)

#define K_LOG2_10 3.3219280948873623f
#define K_LOG10_2 0.30102999566398120f
#define K_ALPHA   0.95f
#define K_1MA     0.05f
#define K_EPS     1e-8f
#define K_TH      0.031622776601683794f   // 10^(-30/20)
// gain' = gain*cm + cb  (cm,cb picked by mask; sign of target-gain is implied by target)
#define K_ATT_CM  0.4493289641172216f     // exp(-0.8)           (attack, target=0.1)
#define K_ATT_CB  0.05506710358827784f    // 0.1*(1-exp(-0.8))
#define K_REL_CM  0.9231163463866358f     // exp(-0.08)          (release, target=1.0)
#define K_REL_CB  0.07688365361336424f    // 1-exp(-0.08)

typedef unsigned int u32;
typedef unsigned long long u64;
typedef u32 v4u __attribute__((ext_vector_type(4)));
typedef int  v4i __attribute__((ext_vector_type(4)));
typedef int  v8i __attribute__((ext_vector_type(8)));

// DPP lane shift within 16-lane rows (halo design keeps all traffic intra-row).
// ROW_SHR n: lane i <- lane i-n  (== shfl_up);  ROW_SHL n: lane i <- lane i+n.
template <int CTRL>
__device__ __forceinline__ float dppf(float v) {
    return __int_as_float(__builtin_amdgcn_update_dpp(
        0, __float_as_int(v), CTRL, 0xF, 0xF, true));
}
#define DPP_ROW_SHR1 0x111
#define DPP_ROW_SHR2 0x112
#define DPP_ROW_SHL1 0x101
#define DPP_ROW_SHL2 0x102

__global__ __launch_bounds__(32)
void howl_kernel(const float* __restrict__ x, float* __restrict__ out)
{
    __shared__ float sIn0[ROWS_IN * ROWF];
    __shared__ float sIn1[ROWS_IN * ROWF];
    __shared__ float sOut[32 * ROWF];

    const int lane  = threadIdx.x;
    const int half  = lane >> 4;              // which 16-lane half
    const int blk   = blockIdx.x;
    const int b     = blk / NBLK_F;
    const int chunk = blk - b * NBLK_F;
    const int f0    = chunk * OUTB;           // first output bin of this chunk
    const int fFirst = f0 - 2;                // first (halo) bin row of the tile
    const int readIdx = lane - 4 * half;      // LDS row this lane consumes (0..27)
    const int f     = fFirst + readIdx;       // this lane's frequency bin
    const bool valid = (f >= 0) && (f < NF);
    const bool band  = (f >= 32) && (f < 256);

    // clip the 28-row span against [0, NF)
    const int rowStart = (chunk == 0) ? 2 : 0;
    const int rowEnd   = min(ROWS_IN, NF - fFirst);   // row r valid iff r < rowEnd
    const int rowCnt   = rowEnd - rowStart;

    const u64 gRowBase = (u64)(uintptr_t)x +
                         4ull * ((u64)(b * NF + fFirst + rowStart) * (u64)NT_T);
    const u32 ldsIn0 = (u32)(uintptr_t)sIn0 + (u32)rowStart * (ROWF * 4);
    const u32 ldsIn1 = (u32)(uintptr_t)sIn1 + (u32)rowStart * (ROWF * 4);

    // D# group1 word0: data_size=4B (code 2), pad_enable, pad_interval=128dw (code 6),
    // pad_amount=4dw (code 3) -> TDM writes LDS rows with stride 132 floats
    const u32 g1d0 = (2u << 16) | (1u << 20) | (6u << 22) | (3u << 25);

    auto issueTile = [&](int t0, int bufsel) {
        const u64 ga = gRowBase + 4ull * (u32)t0;    // tile start (row fFirst+rowStart, col t0)
        v4u g0;
        g0.x = 1u;                                   // count=1 valid descriptor
        g0.y = bufsel ? ldsIn1 : ldsIn0;             // lds_addr (bytes)
        g0.z = (u32)ga;                              // global_addr[31:0]
        g0.w = ((u32)(ga >> 32) & 0x01FFFFFFu) | (2u << 30);  // addr[56:32] | type=2
        v8i g1;
        g1[0] = (int)g1d0;
        g1[1] = (int)((u32)(NT_T - t0) << 16);       // tensor_dim0 (cols remaining) lo16
        g1[2] = (int)((u32)rowCnt << 16);            // dim0 hi16=0 | tensor_dim1 lo16
        g1[3] = (int)(128u << 16);                   // dim1 hi16=0 | tile_dim0=128
        g1[4] = (int)(u32)rowCnt;                    // tile_dim1 | tile_dim2=0
        g1[5] = (int)(u32)NT_T;                      // tensor_dim0_stride = 4000 elems
        g1[6] = 0;
        g1[7] = 0;
        v4i z4 = {0, 0, 0, 0};
#if __clang_major__ >= 23
        v8i z8 = {0, 0, 0, 0, 0, 0, 0, 0};
        __builtin_amdgcn_tensor_load_to_lds(g0, g1, z4, z4, z8, 0);
#else
        __builtin_amdgcn_tensor_load_to_lds(g0, g1, z4, z4, 0);
#endif
    };

    issueTile(0, 0);                 // prefetch tile 0

    float bg = 0.0f, gain = 1.0f;

    for (int j = 0; j < NTILE; ++j) {
        const int t0 = j * TT;
        const int nstep = min(TT, NT_T - t0);   // 128, or 32 on the last tile

        if (j + 1 < NTILE) {
            // all LDS reads of the buffer being refilled must have drained
            asm volatile("s_wait_dscnt 0" ::: "memory");
            issueTile((j + 1) * TT, (j + 1) & 1);
            __builtin_amdgcn_s_wait_tensorcnt(1);   // tile j complete (in-order)
        } else {
            __builtin_amdgcn_s_wait_tensorcnt(0);
        }
        asm volatile("" ::: "memory");              // compiler fence across the wait

        const float* irow = ((j & 1) ? sIn1 : sIn0) + readIdx * ROWF;
        float*       srow = sOut + lane * ROWF;

        for (int tc = 0; tc < nstep; tc += 4) {
            const bool isFirst = (j == 0) && (tc == 0);
            float4 xi = *(const float4*)(irow + tc);
            float xv[4] = {xi.x, xi.y, xi.z, xi.w};

            // frames + background EMA + snr (bg chain is 1 fma/step)
            float fr[4], snr[4];
            #pragma unroll
            for (int k = 0; k < 4; ++k) {
                float xx = valid ? xv[k] : -__builtin_inff();   // dead lanes -> frame 0
                fr[k] = __builtin_amdgcn_exp2f(xx * K_LOG2_10); // 10^x
            }
            #pragma unroll
            for (int k = 0; k < 4; ++k) {
                float bgn = __builtin_fmaf(K_ALPHA, bg, K_1MA * fr[k]);
                bg = (isFirst && k == 0) ? fr[0] : bgn;         // t=0: bg := frame
                snr[k] = fr[k] * __builtin_amdgcn_rcpf(bg + K_EPS);
            }

            // neighbor exchange: pure DPP (intra-16-lane-row thanks to split halo)
            float nm1[4], nm2[4], np1[4], np2[4];
            #pragma unroll
            for (int k = 0; k < 4; ++k) {
                nm1[k] = dppf<DPP_ROW_SHR1>(snr[k]);
                nm2[k] = dppf<DPP_ROW_SHR2>(snr[k]);
                np1[k] = dppf<DPP_ROW_SHL1>(snr[k]);
                np2[k] = dppf<DPP_ROW_SHL2>(snr[k]);
            }

            // mask / gain recurrence (1 fma/step in the chain) / output
            float o[4];
            #pragma unroll
            for (int k = 0; k < 4; ++k) {
                float mx = fmaxf(fmaxf(nm1[k], nm2[k]), fmaxf(np1[k], np2[k]));
                bool mask = (mx < snr[k]) && (snr[k] > K_TH) && band;
                float cm = mask ? K_ATT_CM : K_REL_CM;
                float cb = mask ? K_ATT_CB : K_REL_CB;
                float g2 = __builtin_fmaf(gain, cm, cb);
                gain = (isFirst && k == 0) ? 1.0f : g2;         // t=0: gain := 1
                o[k] = __builtin_amdgcn_logf(__builtin_fmaf(fr[k], gain, K_EPS))
                       * K_LOG10_2;
            }
            *(float4*)(srow + tc) = make_float4(o[0], o[1], o[2], o[3]);
        }

        asm volatile("" ::: "memory");   // single wave: LDS is in-order, fence compiler

        // coalesced writeback: output bin f0+rr lives in lane-row (rr<12 ? rr+2 : rr+6)
        for (int rr = 0; rr < OUTB; ++rr) {
            const int fo = f0 + rr;
            if (fo >= NF) break;
            const int r = (rr < 12) ? (rr + 2) : (rr + 6);
            float4 v = *(const float4*)(sOut + r * ROWF + 4 * lane);
            if (4 * lane < nstep) {
                size_t off = (size_t)(b * NF + fo) * (size_t)NT_T
                           + (size_t)t0 + (size_t)(4 * lane);
                *(float4*)(out + off) = v;
            }
        }
        asm volatile("" ::: "memory");
    }
}

extern "C" void kernel_launch(void* const* d_in, const int* in_sizes, int n_in,
                              void* d_out, int out_size, void* d_ws, size_t ws_size,
                              hipStream_t stream) {
    (void)n_in; (void)out_size; (void)d_ws; (void)ws_size;
    const float* x = (const float*)d_in[0];
    float* out = (float*)d_out;
    const int B = in_sizes[0] / (NF * NT_T);       // 16 for the reference shapes
    dim3 grid(B * NBLK_F), block(32);
    hipLaunchKernelGGL(howl_kernel, grid, block, 0, stream, x, out);
}